// GraphSage_65893388255520
// MI455X (gfx1250) — compile-verified
//
#include <hip/hip_runtime.h>

// ---------------------------------------------------------------------------
// GraphSAGE on MI455X (gfx1250, wave32).
// Layer pattern:  x = [ self | mean(neigh) ]  (16 x 256 per block, in LDS)
//                 out_tile = x @ W            (256x128, fp32 WMMA 16x16x4)
// fp32 end-to-end for numerical parity with the f32 JAX reference; the kernel
// is L2/gather bound (~110-160 MB HBM-side vs 7.3 GFLOP), so fp32 WMMA costs
// nothing versus bf16 and removes quantization risk.
// ---------------------------------------------------------------------------

typedef __attribute__((ext_vector_type(2))) float v2f;
typedef __attribute__((ext_vector_type(4))) float v4f;
typedef __attribute__((ext_vector_type(8))) float v8f;

#define FDIM    128   // feature / hidden / output width
#define DEG     16    // neighbors per node
#define KDIM    256   // concat width = 2*FDIM
#define XSTRIDE 260   // 256 + 4 pad: A-frag ds_load_b64 banks 4m+4k+{0..3}, conflict-free

__global__ __launch_bounds__(256)
void sage_layer(const float* __restrict__ src,    // gather/self source [M, 128]
                const float* __restrict__ wmat,   // [256, 128] row-major
                const int*   __restrict__ nbr,    // [N, 16]
                const int*   __restrict__ nodes,  // nullable: row -> node map
                float*       __restrict__ out)    // [rows, 128]
{
    __shared__ float xs[16 * XSTRIDE];            // 16 rows x 256 (+pad) concat input
    __shared__ int   snbr[16][16];

    const int tid  = threadIdx.x;
    const int row0 = blockIdx.x * 16;
    const int r    = tid >> 4;                    // row within tile (0..15)
    const int j    = tid & 15;                    // 16 threads per row

    const int node = nodes ? nodes[row0 + r] : (row0 + r);
    snbr[r][j] = nbr[node * DEG + j];
    __syncthreads();

    // ---- phase 1: gather + mean + concat into LDS -------------------------
    const int cs = j * 8;                         // 8 columns per thread
    const float* selfp = src + (size_t)node * FDIM + cs;
    v4f s0 = *(const v4f*)(selfp);
    v4f s1 = *(const v4f*)(selfp + 4);

    v4f a0 = (v4f)0.0f, a1 = (v4f)0.0f;
    #pragma unroll
    for (int t = 0; t < DEG; ++t) {
        const float* p = src + (size_t)snbr[r][t] * FDIM + cs;
        a0 += *(const v4f*)(p);
        a1 += *(const v4f*)(p + 4);
    }
    const float inv = 1.0f / (float)DEG;

    float* xr = &xs[r * XSTRIDE];
    *(v4f*)(xr + cs)           = s0;              // concat: self first ...
    *(v4f*)(xr + cs + 4)       = s1;
    *(v4f*)(xr + FDIM + cs)     = a0 * inv;       // ... then neighbor mean
    *(v4f*)(xr + FDIM + cs + 4) = a1 * inv;
    __syncthreads();

    // ---- phase 2: (16x256) @ (256x128) via v_wmma_f32_16x16x4_f32 ---------
    // Each wave owns one 16-wide output panel; 64 WMMAs walk K=256.
    const int lane = tid & 31;
    const int wv   = tid >> 5;                    // wave id 0..7 -> panel
    const int half = lane >> 4;                   // A/B: K pair select; C/D: row+8
    const int mm   = lane & 15;                   // A: row M; B/D: col N

    // A frag (16x4 f32): lane(half,m): VGPR0=A[m][2h], VGPR1=A[m][2h+1]
    const float* xa = &xs[mm * XSTRIDE + 2 * half];
    // B frag (4x16 f32): lane(half,n): VGPR0=B[2h][n], VGPR1=B[2h+1][n]
    const float* wb = wmat + (2 * half) * FDIM + (wv * 16 + mm);

    v8f acc = (v8f)0.0f;
    #pragma unroll 4
    for (int kk = 0; kk < KDIM / 4; ++kk) {
        v2f a = *(const v2f*)(xa + 4 * kk);       // ds_load_b64, bank-clean
        v2f b;
        b.x = wb[(4 * kk) * FDIM];
        b.y = wb[(4 * kk) * FDIM + FDIM];
        acc = __builtin_amdgcn_wmma_f32_16x16x4_f32(
                  /*neg_a=*/false, a, /*neg_b=*/false, b,
                  /*c_mod=*/(short)0, acc, /*reuse_a=*/false, /*reuse_b=*/false);
    }

    // D layout: VGPR v -> row (v + 8*half), col = lane&15 within panel
    float* op = out + (size_t)(row0 + 8 * half) * FDIM + (wv * 16 + mm);
    #pragma unroll
    for (int v = 0; v < 8; ++v)
        op[(size_t)v * FDIM] = acc[v];
}

extern "C" void kernel_launch(void* const* d_in, const int* in_sizes, int n_in,
                              void* d_out, int out_size, void* d_ws, size_t ws_size,
                              hipStream_t stream) {
    (void)in_sizes; (void)n_in; (void)out_size; (void)ws_size;

    const float* features     = (const float*)d_in[0];  // [N, 128]
    const float* w1           = (const float*)d_in[1];  // [256, 128]
    const float* w2           = (const float*)d_in[2];  // [256, 128]
    const int*   neighbor_idx = (const int*)  d_in[3];  // [N, 16]
    const int*   nodes        = (const int*)  d_in[4];  // [B]

    float* out = (float*)d_out;                         // [8192, 128]
    float* h1  = (float*)d_ws;                          // [N, 128] = 51.2 MB scratch

    const int N = 100000;                               // multiple of 16
    const int B = 8192;                                 // multiple of 16

    // Layer 1: h1 = [feat | mean(neigh feat)] @ w1, all N nodes.
    sage_layer<<<N / 16, 256, 0, stream>>>(features, w1, neighbor_idx, nullptr, h1);
    // Layer 2: out = [h1[nodes] | mean(h1[neigh(nodes)])] @ w2, batch nodes.
    sage_layer<<<B / 16, 256, 0, stream>>>(h1, w2, neighbor_idx, nodes, out);
}